// BERT_CRF_NER_18605798326646
// MI455X (gfx1250) — compile-verified
//
#include <hip/hip_runtime.h>
#include <hip/hip_bf16.h>

#define BB 64
#define SS 512
#define DD 768
#define NL 10

typedef float v2f __attribute__((ext_vector_type(2)));
typedef float v8f __attribute__((ext_vector_type(8)));

// ---------------------------------------------------------------------------
// Kernel A: per-row segmentation. One block (512 threads) per batch row.
// Produces: Lr[b] (valid length), nw[b] (#words), startpos[b*SS + w] (token
// index where word w starts).
// ---------------------------------------------------------------------------
__global__ __launch_bounds__(SS) void seg_kernel(const int* __restrict__ mask,
                                                 const int* __restrict__ flm,
                                                 int* __restrict__ Lr,
                                                 int* __restrict__ nw,
                                                 int* __restrict__ startpos) {
    __shared__ int sm[SS];
    const int b = blockIdx.x;
    const int s = threadIdx.x;
    const int m = mask[b * SS + s];

    // valid length L = sum(mask) via tree reduction
    sm[s] = m;
    __syncthreads();
    for (int off = SS / 2; off > 0; off >>= 1) {
        if (s < off) sm[s] += sm[s + off];
        __syncthreads();
    }
    const int L = sm[0];
    __syncthreads();

    // inner tokens are strictly between [CLS] (pos 0) and [SEP] (pos L-1)
    const int inner = (m > 0) && (s >= 1) && (s <= L - 2);
    const int start = inner && (flm[b * SS + s] > 0);

    // inclusive scan of word-start flags (Hillis-Steele in LDS)
    sm[s] = start;
    __syncthreads();
    for (int off = 1; off < SS; off <<= 1) {
        int v = sm[s];
        int add = (s >= off) ? sm[s - off] : 0;
        __syncthreads();
        sm[s] = v + add;
        __syncthreads();
    }
    const int incl = sm[s];

    if (start) startpos[b * SS + (incl - 1)] = s;
    if (s == 0) Lr[b] = L;
    if (s == SS - 1) nw[b] = incl;
}

// ---------------------------------------------------------------------------
// Kernel B: per-token projection proj[t,l] = dot(feat[t,:], W[l,:]) using
// V_WMMA_F32_16X16X4_F32. One wave per 16-token tile, 8 waves per block.
// W (padded to 16 label rows, rows 10..15 = 0) staged in LDS as Wl[k*16+n].
// K slots are permuted consistently between A and B fragments so each lane's
// A data comes from one global_load_b128 of 4 consecutive K values.
// ---------------------------------------------------------------------------
__global__ __launch_bounds__(256) void proj_kernel(const float* __restrict__ feat,
                                                   const float* __restrict__ W,
                                                   float* __restrict__ proj) {
    __shared__ float Wl[DD * 16];
    const int tid = threadIdx.x;
    for (int i = tid; i < DD * 16; i += 256) {
        const int k = i >> 4;
        const int n = i & 15;
        Wl[i] = (n < NL) ? W[n * DD + k] : 0.0f;
    }
    __syncthreads();

    const int wave = tid >> 5;
    const int lane = tid & 31;
    const int tile = blockIdx.x * 8 + wave;          // 16 tokens per tile
    const int mrow = lane & 15;                      // A-matrix M for this lane
    const int h    = lane >> 4;                      // lane half selects K pair
    const int n    = lane & 15;                      // B/D-matrix N for this lane

    const float* fp = feat + (size_t)(tile * 16 + mrow) * DD;

    v8f acc = {};
    for (int k0 = 0; k0 < DD; k0 += 8) {
        // 4 consecutive K values for this lane's half (one b128 load)
        const float4 f = *(const float4*)(fp + k0 + 4 * h);
        const int kb = k0 + 4 * h;                   // same K base for A and B

        v2f a1 = {f.x, f.y};
        v2f b1 = {Wl[kb * 16 + n], Wl[(kb + 1) * 16 + n]};
        acc = __builtin_amdgcn_wmma_f32_16x16x4_f32(
            false, a1, false, b1, (short)0, acc, false, false);

        v2f a2 = {f.z, f.w};
        v2f b2 = {Wl[(kb + 2) * 16 + n], Wl[(kb + 3) * 16 + n]};
        acc = __builtin_amdgcn_wmma_f32_16x16x4_f32(
            false, a2, false, b2, (short)0, acc, false, false);
    }

    // D layout: VGPR v -> M = v + 8*(lane/16), N = lane%16
    if (n < NL) {
        const int mbase = tile * 16 + h * 8;
#pragma unroll
        for (int v = 0; v < 8; ++v) {
            proj[(size_t)(mbase + v) * NL + n] = acc[v];
        }
    }
}

// ---------------------------------------------------------------------------
// Kernel C: per (word-slot, label) mean-pool of projections + bias + sigmoid.
// Words are contiguous token runs [startpos[w], startpos[w+1]) (last word ends
// at L-1 exclusive). Empty slots produce sigmoid(bias) like the reference.
// ---------------------------------------------------------------------------
__global__ __launch_bounds__(256) void pool_kernel(const float* __restrict__ proj,
                                                   const int* __restrict__ Lr,
                                                   const int* __restrict__ nw,
                                                   const int* __restrict__ startpos,
                                                   const float* __restrict__ bias,
                                                   float* __restrict__ out) {
    const int tid = blockIdx.x * blockDim.x + threadIdx.x;
    const int l = tid & 15;
    const int slot = tid >> 4;
    if (slot >= BB * SS || l >= NL) return;

    const int b = slot / SS;
    const int s = slot % SS;
    const int w = nw[b];

    float x = 0.0f;
    if (s < w) {
        const int t0 = startpos[b * SS + s];
        const int t1 = (s + 1 < w) ? startpos[b * SS + s + 1] : (Lr[b] - 1);
        float sum = 0.0f;
        for (int t = t0; t < t1; ++t) {
            sum += proj[(size_t)(b * SS + t) * NL + l];
        }
        x = sum / (float)(t1 - t0);
    }
    x += bias[l];
    out[(size_t)slot * NL + l] = 1.0f / (1.0f + __expf(-x));
}

// ---------------------------------------------------------------------------
// Launch
// ---------------------------------------------------------------------------
extern "C" void kernel_launch(void* const* d_in, const int* in_sizes, int n_in,
                              void* d_out, int out_size, void* d_ws, size_t ws_size,
                              hipStream_t stream) {
    const float* feat = (const float*)d_in[0];   // [B,S,D] f32
    const int*   mask = (const int*)d_in[1];     // [B,S] i32
    const int*   flm  = (const int*)d_in[2];     // [B,S] i32
    const float* W    = (const float*)d_in[3];   // [10,768] f32
    const float* bias = (const float*)d_in[4];   // [10] f32
    float* out = (float*)d_out;                  // [B,S,10] f32

    // workspace layout
    float* proj   = (float*)d_ws;                      // B*S*NL floats
    int*   Lr     = (int*)(proj + (size_t)BB * SS * NL);
    int*   nw     = Lr + BB;
    int*   sp     = nw + BB;                           // B*S ints

    seg_kernel<<<BB, SS, 0, stream>>>(mask, flm, Lr, nw, sp);

    // 32768 tokens / 16 per tile / 8 tiles per block = 256 blocks
    proj_kernel<<<(BB * SS) / (16 * 8), 256, 0, stream>>>(feat, W, proj);

    const int total = BB * SS * 16;
    pool_kernel<<<(total + 255) / 256, 256, 0, stream>>>(proj, Lr, nw, sp, bias, out);
}